// SimGCN_88064009437430
// MI455X (gfx1250) — compile-verified
//
#include <hip/hip_runtime.h>

typedef __attribute__((ext_vector_type(2))) float v2f;
typedef __attribute__((ext_vector_type(8))) float v8f;

#define D 64

// ---------------------------------------------------------------------------
// out[0..255] = 0
__global__ __launch_bounds__(256) void k_zero_out(float* __restrict__ out) {
    out[threadIdx.x] = 0.0f;
}

// deg[n] = 1.0 (self-loop contribution)
__global__ __launch_bounds__(256) void k_deg_init(float* __restrict__ deg, int n) {
    int i = blockIdx.x * 256 + threadIdx.x;
    if (i < n) deg[i] = 1.0f;
}

// deg[dst[e]] += 1
__global__ __launch_bounds__(256) void k_deg_accum(const int* __restrict__ dst,
                                                   float* __restrict__ deg, int nE) {
    int e = blockIdx.x * 256 + threadIdx.x;
    if (e < nE) atomicAdd(&deg[dst[e]], 1.0f);
}

// dinv[n] = rsqrt(deg[n])   (deg >= 1 always, thanks to self loops)
__global__ __launch_bounds__(256) void k_deg_finalize(float* __restrict__ deg, int n) {
    int i = blockIdx.x * 256 + threadIdx.x;
    if (i < n) deg[i] = rsqrtf(deg[i]);
}

// agg[n*64+f] = b[f]
__global__ __launch_bounds__(256) void k_agg_init(float* __restrict__ agg,
                                                  const float* __restrict__ b,
                                                  long long total) {
    long long i = (long long)blockIdx.x * 256 + threadIdx.x;
    if (i < total) agg[i] = b[i & (D - 1)];
}

// ---------------------------------------------------------------------------
// H = X * W   (X: nRows x 64, W: 64 x 64, H: nRows x 64), fp32 WMMA path.
// One wave (32 lanes) computes a 16-row x 64-col strip of H.
// A tile 16x4 f32: lane L<16 holds {A[M=L][K=k], A[M=L][K=k+1]},
//                  lane L>=16 holds {A[M=L-16][K=k+2], A[M=L-16][K=k+3]}.
// B tile 4x16 f32: symmetric (col = L&15, same K pair).
// C/D 16x16 f32:  VGPR v, lane L -> row v + 8*(L>>4), col L&15.
__global__ __launch_bounds__(128) void k_gemm_wmma(const float* __restrict__ X,
                                                   const float* __restrict__ W,
                                                   float* __restrict__ H, int nRows) {
    const int lane  = threadIdx.x & 31;
    const int wave  = threadIdx.x >> 5;
    const int strip = blockIdx.x * 4 + wave;      // 16-row strip index (wave-uniform)
    if (strip * 16 >= nRows) return;              // whole-wave exit: EXEC stays all-1s
    const int m0 = strip * 16;
    const int mr = lane & 15;                     // A-row / B-col / D-col within tile
    const int kh = (lane >> 4) << 1;              // 0 for lanes 0-15, 2 for lanes 16-31

    v8f acc0 = {}, acc1 = {}, acc2 = {}, acc3 = {};
    const float* xrow = X + (size_t)(m0 + mr) * D;

    #pragma unroll
    for (int k = 0; k < D; k += 4) {
        v2f a;
        a.x = xrow[k + kh];
        a.y = xrow[k + kh + 1];
        const float* w0 = W + (size_t)(k + kh) * D;      // row K=k+kh   of W
        const float* w1 = W + (size_t)(k + kh + 1) * D;  // row K=k+kh+1 of W
        v2f b0 = { w0[mr],      w1[mr]      };
        v2f b1 = { w0[16 + mr], w1[16 + mr] };
        v2f b2 = { w0[32 + mr], w1[32 + mr] };
        v2f b3 = { w0[48 + mr], w1[48 + mr] };
        acc0 = __builtin_amdgcn_wmma_f32_16x16x4_f32(false, a, false, b0, (short)0, acc0, false, false);
        acc1 = __builtin_amdgcn_wmma_f32_16x16x4_f32(false, a, false, b1, (short)0, acc1, false, false);
        acc2 = __builtin_amdgcn_wmma_f32_16x16x4_f32(false, a, false, b2, (short)0, acc2, false, false);
        acc3 = __builtin_amdgcn_wmma_f32_16x16x4_f32(false, a, false, b3, (short)0, acc3, false, false);
    }

    const int rbase = m0 + ((lane >> 4) << 3);
    #pragma unroll
    for (int v = 0; v < 8; ++v) {
        float* dst = H + (size_t)(rbase + v) * D + mr;
        dst[0]  = acc0[v];
        dst[16] = acc1[v];
        dst[32] = acc2[v];
        dst[48] = acc3[v];
    }
}

// ---------------------------------------------------------------------------
// agg[col[e]][f] += H[row[e]][f] * dinv[row]*dinv[col]
// e in [0, nE)        : real edges (src/dst arrays)
// e in [nE, nE+nN)    : self loops, norm = dinv^2 = 1/deg
__global__ __launch_bounds__(256) void k_scatter(const float* __restrict__ H,
                                                 const int* __restrict__ src,
                                                 const int* __restrict__ dst,
                                                 const float* __restrict__ dinv,
                                                 float* __restrict__ agg,
                                                 int nE, int nN) {
    long long idx   = (long long)blockIdx.x * 256 + threadIdx.x;
    long long total = (long long)(nE + nN) * D;
    if (idx >= total) return;
    int e = (int)(idx >> 6);
    int f = (int)(idx & (D - 1));
    int r, c; float nrm;
    if (e < nE) {
        r = src[e]; c = dst[e];
        nrm = dinv[r] * dinv[c];
    } else {
        r = c = e - nE;
        float d = dinv[r];
        nrm = d * d;
    }
    atomicAdd(&agg[(size_t)c * D + f], H[(size_t)r * D + f] * nrm);
}

// ---------------------------------------------------------------------------
// out[layer*64 + f] += (1/N) * sum_n agg[n][f]
__global__ __launch_bounds__(256) void k_colmean(const float* __restrict__ agg,
                                                 float* __restrict__ out,
                                                 int nN, int layer) {
    int f   = threadIdx.x & (D - 1);
    int sub = threadIdx.x >> 6;            // 0..3
    int stride = gridDim.x * 4;
    float s = 0.0f;
    for (int n = blockIdx.x * 4 + sub; n < nN; n += stride)
        s += agg[(size_t)n * D + f];
    atomicAdd(&out[layer * D + f], s * (1.0f / (float)nN));
}

// ---------------------------------------------------------------------------
extern "C" void kernel_launch(void* const* d_in, const int* in_sizes, int n_in,
                              void* d_out, int out_size, void* d_ws, size_t ws_size,
                              hipStream_t stream) {
    const float* x   = (const float*)d_in[0];
    const int*   ei  = (const int*)d_in[1];
    const int    N   = in_sizes[0] / D;       // 50000
    const int    E   = in_sizes[1] / 2;       // 800000
    const int*   src = ei;                    // edge_index[0] = source
    const int*   dst = ei + E;                // edge_index[1] = target
    const float* Ws[4] = { (const float*)d_in[2], (const float*)d_in[4],
                           (const float*)d_in[6], (const float*)d_in[8] };
    const float* Bs[4] = { (const float*)d_in[3], (const float*)d_in[5],
                           (const float*)d_in[7], (const float*)d_in[9] };
    float* out = (float*)d_out;

    // Workspace layout (floats): dinv[N] | H[N*D] | xA[N*D] | xB[N*D]
    float* dinv = (float*)d_ws;
    float* H    = dinv + N;
    float* xA   = H + (size_t)N * D;
    float* xB   = xA + (size_t)N * D;

    const int TB = 256;

    k_zero_out<<<1, 256, 0, stream>>>(out);

    // symmetric normalization coefficients
    k_deg_init    <<<(N + TB - 1) / TB, TB, 0, stream>>>(dinv, N);
    k_deg_accum   <<<(E + TB - 1) / TB, TB, 0, stream>>>(dst, dinv, E);
    k_deg_finalize<<<(N + TB - 1) / TB, TB, 0, stream>>>(dinv, N);

    const long long aggElems  = (long long)N * D;
    const long long scatElems = (long long)(E + N) * D;
    const int aggBlocks  = (int)((aggElems  + TB - 1) / TB);
    const int scatBlocks = (int)((scatElems + TB - 1) / TB);
    const int strips     = (N + 15) / 16;
    const int gemmBlocks = (strips + 3) / 4;   // 4 waves (16-row strips) per block

    const float* cur = x;
    float* pingpong[2] = { xA, xB };
    for (int l = 0; l < 4; ++l) {
        float* agg = pingpong[l & 1];
        k_gemm_wmma<<<gemmBlocks, 128, 0, stream>>>(cur, Ws[l], H, N);
        k_agg_init <<<aggBlocks, TB, 0, stream>>>(agg, Bs[l], aggElems);
        k_scatter  <<<scatBlocks, TB, 0, stream>>>(H, src, dst, dinv, agg, E, N);
        k_colmean  <<<256, TB, 0, stream>>>(agg, out, N, l);
        cur = agg;
    }
}